// Customlosskll1_11098195493093
// MI455X (gfx1250) — compile-verified
//
#include <hip/hip_runtime.h>
#include <math.h>

#define EPS 1e-6f

typedef __attribute__((ext_vector_type(2))) float v2f;
typedef __attribute__((ext_vector_type(8))) float v8f;

// ---- orderable-uint encoding for float atomic min/max ----
__device__ __forceinline__ unsigned fkey(float f) {
    unsigned u = __float_as_uint(f);
    return (u & 0x80000000u) ? ~u : (u | 0x80000000u);
}
__device__ __forceinline__ float keyf(unsigned k) {
    return (k & 0x80000000u) ? __uint_as_float(k & 0x7fffffffu)
                             : __uint_as_float(~k);
}

// workspace layout (as float/uint words):
//   [0]            partA sum (float, atomicAdd)
//   [1],[2]        inputo min/max keys (uint)
//   [3],[4]        target min/max keys (uint)
//   [16 ..)        hist_pred (B*bins floats), then hist_gt (B*bins floats)

__global__ void k_init(float* wsf, unsigned* wsu, int histTotal) {
    int i = blockIdx.x * blockDim.x + threadIdx.x;
    if (i == 0) {
        wsf[0] = 0.0f;
        wsu[1] = 0xFFFFFFFFu; wsu[2] = 0u;   // inputo min/max keys
        wsu[3] = 0xFFFFFFFFu; wsu[4] = 0u;   // target min/max keys
    }
    for (int j = i; j < histTotal; j += gridDim.x * blockDim.x)
        wsf[16 + j] = 0.0f;
}

// ---- pass 1: fused partA + global min/max of inputo and target ----
__global__ __launch_bounds__(256)
void k_parta_minmax(const float* __restrict__ in, const float* __restrict__ tg,
                    const float* __restrict__ w1, float* wsf, unsigned* wsu,
                    long long N) {
    const long long N4 = N >> 2;
    const long long stride = (long long)gridDim.x * blockDim.x;
    long long gtid = (long long)blockIdx.x * blockDim.x + threadIdx.x;

    float acc = 0.0f;
    float imn = INFINITY, imx = -INFINITY, tmn = INFINITY, tmx = -INFINITY;

    for (long long i = gtid; i < N4; i += stride) {
        float4 a = ((const float4*)in)[i];
        float4 b = ((const float4*)tg)[i];
        float4 w = ((const float4*)w1)[i];
        __builtin_prefetch(&((const float4*)in)[i + stride], 0, 0);
        __builtin_prefetch(&((const float4*)tg)[i + stride], 0, 0);
        __builtin_prefetch(&((const float4*)w1)[i + stride], 0, 0);
        {
            float l = fabsf(a.x - b.x), ww = w.x + EPS;
            acc += l * ww + l / ww;
            imn = fminf(imn, a.x); imx = fmaxf(imx, a.x);
            tmn = fminf(tmn, b.x); tmx = fmaxf(tmx, b.x);
        }
        {
            float l = fabsf(a.y - b.y), ww = w.y + EPS;
            acc += l * ww + l / ww;
            imn = fminf(imn, a.y); imx = fmaxf(imx, a.y);
            tmn = fminf(tmn, b.y); tmx = fmaxf(tmx, b.y);
        }
        {
            float l = fabsf(a.z - b.z), ww = w.z + EPS;
            acc += l * ww + l / ww;
            imn = fminf(imn, a.z); imx = fmaxf(imx, a.z);
            tmn = fminf(tmn, b.z); tmx = fmaxf(tmx, b.z);
        }
        {
            float l = fabsf(a.w - b.w), ww = w.w + EPS;
            acc += l * ww + l / ww;
            imn = fminf(imn, a.w); imx = fmaxf(imx, a.w);
            tmn = fminf(tmn, b.w); tmx = fmaxf(tmx, b.w);
        }
    }
    // scalar tail
    for (long long i = (N4 << 2) + gtid; i < N; i += stride) {
        float a = in[i], b = tg[i];
        float l = fabsf(a - b), ww = w1[i] + EPS;
        acc += l * ww + l / ww;
        imn = fminf(imn, a); imx = fmaxf(imx, a);
        tmn = fminf(tmn, b); tmx = fmaxf(tmx, b);
    }

    // wave32 reduce
    for (int off = 16; off >= 1; off >>= 1) {
        acc += __shfl_down(acc, off, 32);
        imn = fminf(imn, __shfl_down(imn, off, 32));
        imx = fmaxf(imx, __shfl_down(imx, off, 32));
        tmn = fminf(tmn, __shfl_down(tmn, off, 32));
        tmx = fmaxf(tmx, __shfl_down(tmx, off, 32));
    }
    __shared__ float sa[8], s0[8], s1[8], s2[8], s3[8];
    int lane = threadIdx.x & 31, wid = threadIdx.x >> 5;
    if (lane == 0) { sa[wid] = acc; s0[wid] = imn; s1[wid] = imx; s2[wid] = tmn; s3[wid] = tmx; }
    __syncthreads();
    if (threadIdx.x == 0) {
        float A = 0.0f, m0 = INFINITY, M0 = -INFINITY, m1 = INFINITY, M1 = -INFINITY;
        int nw = (blockDim.x + 31) >> 5;
        for (int w = 0; w < nw; ++w) {
            A += sa[w];
            m0 = fminf(m0, s0[w]); M0 = fmaxf(M0, s1[w]);
            m1 = fminf(m1, s2[w]); M1 = fmaxf(M1, s3[w]);
        }
        atomicAdd(&wsf[0], A);
        atomicMin(&wsu[1], fkey(m0)); atomicMax(&wsu[2], fkey(M0));
        atomicMin(&wsu[3], fkey(m1)); atomicMax(&wsu[4], fkey(M1));
    }
}

// soft-histogram scatter: element x with k = clamp(floor(x*bins)) contributes
// (h_{k+1}-x) to bin k (if 1<=k<=bins-2) and (x-h_k) to bin k+1 (if k<=bins-3)
__device__ __forceinline__ void hacc(float* h, float xn, int bins, float fbins, float delta) {
    int k = (int)floorf(xn * fbins);
    k = k < 0 ? 0 : (k > bins - 1 ? bins - 1 : k);
    float lower = (float)k * delta;
    if (k >= 1 && k <= bins - 2) atomicAdd(&h[k], (lower + delta) - xn);
    if (k <= bins - 3)           atomicAdd(&h[k + 1], xn - lower);
}

// ---- pass 2: per-image LDS-privatized soft histograms (pred & gt) ----
extern __shared__ float sh[];
__global__ __launch_bounds__(256)
void k_hist(const float* __restrict__ in, const float* __restrict__ tg,
            const unsigned* __restrict__ wsu, float* wsf,
            int B, int bins, long long M) {
    float* shP = sh;
    float* shG = sh + bins;
    for (int j = threadIdx.x; j < 2 * bins; j += blockDim.x) sh[j] = 0.0f;
    __syncthreads();

    const int b = blockIdx.y;
    const float* inb = in + (long long)b * M;
    const float* tgb = tg + (long long)b * M;
    const float imn = keyf(wsu[1]), imx = keyf(wsu[2]);
    const float tmn = keyf(wsu[3]), tmx = keyf(wsu[4]);
    const float iinv = 1.0f / (imx - imn);
    const float tinv = 1.0f / (tmx - tmn);
    const float fbins = (float)bins, delta = 1.0f / fbins;

    const long long M4 = M >> 2;
    const long long stride = (long long)gridDim.x * blockDim.x;
    long long gtid = (long long)blockIdx.x * blockDim.x + threadIdx.x;

    for (long long i = gtid; i < M4; i += stride) {
        float4 a = ((const float4*)inb)[i];
        float4 t = ((const float4*)tgb)[i];
        __builtin_prefetch(&((const float4*)inb)[i + stride], 0, 0);
        __builtin_prefetch(&((const float4*)tgb)[i + stride], 0, 0);
        hacc(shP, (a.x - imn) * iinv, bins, fbins, delta);
        hacc(shP, (a.y - imn) * iinv, bins, fbins, delta);
        hacc(shP, (a.z - imn) * iinv, bins, fbins, delta);
        hacc(shP, (a.w - imn) * iinv, bins, fbins, delta);
        hacc(shG, (t.x - tmn) * tinv, bins, fbins, delta);
        hacc(shG, (t.y - tmn) * tinv, bins, fbins, delta);
        hacc(shG, (t.z - tmn) * tinv, bins, fbins, delta);
        hacc(shG, (t.w - tmn) * tinv, bins, fbins, delta);
    }
    for (long long i = (M4 << 2) + gtid; i < M; i += stride) {
        hacc(shP, (inb[i] - imn) * iinv, bins, fbins, delta);
        hacc(shG, (tgb[i] - tmn) * tinv, bins, fbins, delta);
    }
    __syncthreads();

    float* gP = wsf + 16 + (long long)b * bins;
    float* gG = wsf + 16 + (long long)B * bins + (long long)b * bins;
    for (int j = threadIdx.x; j < bins; j += blockDim.x) {
        float p = shP[j], g = shG[j];
        if (p != 0.0f) atomicAdd(&gP[j], p);
        if (g != 0.0f) atomicAdd(&gG[j], g);
    }
}

// ---- pass 3: WMMA-f32 histogram sums + KL reduction + final scalar ----
// Ones-vector trick: with A = all-ones 16x4 f32, V_WMMA_F32_16X16X4_F32 gives
// D[m][n] = C[m][n] + sum_k B[k][n]; so accumulating C across iterations and
// summing row 0 (lanes 0..15, VGPR0 of D) yields the total of all B elements fed.
__global__ __launch_bounds__(256)
void k_final(const float* __restrict__ we2, const float* __restrict__ wsf,
             float* __restrict__ out, int B, int bins, long long Ntot) {
    const int TOT = B * bins;
    const float* hP = wsf + 16;
    const float* hG = hP + TOT;
    __shared__ float sP[8], sG[8], sB[8], S[2];

    const int lane = threadIdx.x & 31, wid = threadIdx.x >> 5;

    // phase 1: each of the 8 waves sums a chunk of both histograms via WMMA
    {
        const int chunk = TOT / 8;
        const int vec = chunk & ~63;           // 64 f32 per wmma issue
        const int base = wid * chunk;
        v2f a; a.x = 1.0f; a.y = 1.0f;         // ones A-matrix (16x4 f32)
        v8f cP = {}; v8f cG = {};
        for (int i = 0; i < vec; i += 64) {
            int idx = base + i + lane * 2;
            v2f bP; bP.x = hP[idx]; bP.y = hP[idx + 1];
            v2f bG; bG.x = hG[idx]; bG.y = hG[idx + 1];
            cP = __builtin_amdgcn_wmma_f32_16x16x4_f32(
                false, a, false, bP, (short)0, cP, false, false);
            cG = __builtin_amdgcn_wmma_f32_16x16x4_f32(
                false, a, false, bG, (short)0, cG, false, false);
        }
        float pv = cP[0];   // row M=0, column N=lane (lanes 0..15)
        float gv = cG[0];
        if (lane < 16) {    // scalar tail handled by the M=0 half only
            for (int i = base + vec + lane; i < base + chunk; i += 16) {
                pv += hP[i]; gv += hG[i];
            }
        }
        for (int off = 8; off >= 1; off >>= 1) {   // reduce lanes 0..15
            pv += __shfl_down(pv, off, 16);
            gv += __shfl_down(gv, off, 16);
        }
        if (lane == 0) { sP[wid] = pv; sG[wid] = gv; }
    }
    __syncthreads();
    if (threadIdx.x == 0) {
        float a = 0.0f, b = 0.0f;
        for (int w = 0; w < 8; ++w) { a += sP[w]; b += sG[w]; }
        S[0] = a; S[1] = b;
    }
    __syncthreads();

    // pdf = hist / sum(hist)   (the 1/delta factors cancel)
    const float invSP = 1.0f / S[0];
    const float invSG = 1.0f / S[1];

    // phase 2: KL + we2 weighting, mean over TOT
    float acc = 0.0f;
    for (int j = threadIdx.x; j < TOT; j += blockDim.x) {
        float p = hP[j] * invSP;
        float g = hG[j] * invSG;
        float kld = fabsf(expf(g) * (g - p));
        float w2 = we2[j] + EPS;               // we2 flat is [B, bins]
        acc += kld * w2 + kld / w2;
    }
    for (int off = 16; off >= 1; off >>= 1) acc += __shfl_down(acc, off, 32);
    if (lane == 0) sB[wid] = acc;
    __syncthreads();
    if (threadIdx.x == 0) {
        float pb = 0.0f;
        for (int w = 0; w < 8; ++w) pb += sB[w];
        float parta = wsf[0] / (float)Ntot;
        float partb = pb / (float)TOT;
        out[0] = 4.0f * parta + partb;
    }
}

extern "C" void kernel_launch(void* const* d_in, const int* in_sizes, int n_in,
                              void* d_out, int out_size, void* d_ws, size_t ws_size,
                              hipStream_t stream) {
    const float* inputo = (const float*)d_in[0];
    const float* target = (const float*)d_in[1];
    const float* we1    = (const float*)d_in[2];
    const float* we2    = (const float*)d_in[3];

    long long N = in_sizes[0];       // B*C*H*W
    int BW = in_sizes[3];            // B*C*W  (we2 flat)
    long long H = N / BW;            // C==1, square image => H==W
    int bins = (int)H;               // bins = W = image_size
    int B = BW / bins;
    long long M = (long long)bins * bins;  // per-image elements

    float*    wsf = (float*)d_ws;
    unsigned* wsu = (unsigned*)d_ws;

    k_init<<<64, 256, 0, stream>>>(wsf, wsu, 2 * B * bins);
    k_parta_minmax<<<2048, 256, 0, stream>>>(inputo, target, we1, wsf, wsu, N);
    dim3 g2(512, (unsigned)B);
    size_t shbytes = (size_t)2 * bins * sizeof(float);
    k_hist<<<g2, 256, shbytes, stream>>>(inputo, target, wsu, wsf, B, bins, M);
    k_final<<<1, 256, 0, stream>>>(we2, wsf, (float*)d_out, B, bins, N);
}